// NNUE_84344567759346
// MI455X (gfx1250) — compile-verified
//
#include <hip/hip_runtime.h>

typedef __attribute__((ext_vector_type(2))) float v2f;
typedef __attribute__((ext_vector_type(8))) float v8f;
typedef int v4i_vs __attribute__((vector_size(16)));   // matches builtin param type

#define EMB_DIM 2048
#define FEAT_L  32
#define TM      16          // batch rows per block
#define CK      64          // K-chunk size
#define KS      (CK + 4)    // padded LDS row stride in floats (272B, 16B aligned, bank-stride 4)
#define NCHUNK  ((2 * EMB_DIM) / CK)   // 64

// ---- CDNA5 async global->LDS (no VGPR transit, ASYNCcnt-tracked) ----
#if defined(__has_builtin)
#if __has_builtin(__builtin_amdgcn_global_load_async_to_lds_b128)
#define ASYNC_LDS_B128(gptr, sptr)                                          \
  __builtin_amdgcn_global_load_async_to_lds_b128(                           \
      (__attribute__((address_space(1))) v4i_vs*)(gptr),                    \
      (__attribute__((address_space(3))) v4i_vs*)(sptr), 0, 0)
#define HAVE_ASYNC_BUILTIN 1
#endif
#if __has_builtin(__builtin_amdgcn_s_wait_asynccnt)
#define WAIT_ASYNC() __builtin_amdgcn_s_wait_asynccnt(0)
#define HAVE_WAIT_BUILTIN 1
#endif
#endif
#ifndef HAVE_ASYNC_BUILTIN
#define ASYNC_LDS_B128(gptr, sptr)                                          \
  asm volatile("global_load_async_to_lds_b128 %0, %1, off"                  \
               :: "v"((unsigned)(size_t)(sptr)), "v"(gptr) : "memory")
#endif
#ifndef HAVE_WAIT_BUILTIN
#define WAIT_ASYNC() asm volatile("s_wait_asynccnt 0x0" ::: "memory")
#endif

struct __align__(16) Smem {
  int   sidx[2][TM][FEAT_L];   // perspective-resolved indices  (4 KB)
  float xs[TM][KS];            // accumulated activation chunk  (4.25 KB)
  float w1s[128][KS];          // W1 chunk                      (34 KB)
  float part[8][32][8];        // layer-2 partials              (8 KB)
};

__global__ __launch_bounds__(256) void nnue_fused_kernel(
    const int* __restrict__ white, const int* __restrict__ black,
    const int* __restrict__ stm,   const float* __restrict__ emb,
    const float* __restrict__ W1,  const float* __restrict__ b1,
    const float* __restrict__ W2,  const float* __restrict__ b2,
    float* __restrict__ out)
{
  __shared__ Smem sm;
  const int t   = threadIdx.x;
  const int gr0 = blockIdx.x * TM;

  // Stage perspective-resolved index lists: first half of x uses (stm ? white : black)
  if (t < TM) {
    const int  s  = stm[gr0 + t];
    const int* fi = s ? white : black;
    const int* se = s ? black : white;
    for (int l = 0; l < FEAT_L; ++l) {
      sm.sidx[0][t][l] = fi[(gr0 + t) * FEAT_L + l];
      sm.sidx[1][t][l] = se[(gr0 + t) * FEAT_L + l];
    }
  }
  __syncthreads();

  const int lane = t & 31;
  const int wv   = t >> 5;          // wave id = N-tile id (0..7)
  const int fm   = lane & 15;       // fragment row / col within tile
  const int hi2  = (lane >> 4) << 1;// K sub-offset: 0 for lanes 0-15, 2 for 16-31

  const int gr = t >> 4;            // gather row (0..15), 16 threads per row
  const int cg = (t & 15) * 4;      // gather col offset within chunk (float4)

  v8f cf = {0.f, 0.f, 0.f, 0.f, 0.f, 0.f, 0.f, 0.f};

  for (int c = 0; c < NCHUNK; ++c) {
    const int half = c >> 5;              // which perspective half of x
    const int cb   = (c & 31) * CK;       // column base within embedding row
    const int kglb = c * CK;              // global K base (into W1 columns)

    // --- gather-accumulate this thread's float4 of the embedding sum ---
    // Two independent accumulator chains; 16 b128 loads in flight per thread.
    float a0x = 0.f, a0y = 0.f, a0z = 0.f, a0w = 0.f;
    float a1x = 0.f, a1y = 0.f, a1z = 0.f, a1w = 0.f;
    #pragma unroll 8
    for (int l = 0; l < FEAT_L; l += 2) {
      const int2  p   = *(const int2*)&sm.sidx[half][gr][l];
      const float m0  = (p.x != 0) ? 1.0f : 0.0f;
      const float m1  = (p.y != 0) ? 1.0f : 0.0f;
      const float4 v0 = *(const float4*)(emb + (size_t)p.x * EMB_DIM + cb + cg);
      const float4 v1 = *(const float4*)(emb + (size_t)p.y * EMB_DIM + cb + cg);
      a0x += v0.x * m0; a0y += v0.y * m0; a0z += v0.z * m0; a0w += v0.w * m0;
      a1x += v1.x * m1; a1y += v1.y * m1; a1z += v1.z * m1; a1w += v1.w * m1;
    }

    __syncthreads();   // previous chunk's WMMA reads complete

    // --- stage W1 chunk (128 x 64) straight into LDS, async, no VGPR transit ---
    #pragma unroll
    for (int i = 0; i < 8; ++i) {
      const int e  = t + i * 256;        // float4 id within 128x16
      const int j  = e >> 4;             // W1 row
      const int c4 = (e & 15) * 4;       // col within chunk
      ASYNC_LDS_B128(W1 + (size_t)j * (2 * EMB_DIM) + kglb + c4,
                     &sm.w1s[j][c4]);
    }

    *(float4*)&sm.xs[gr][cg] =
        make_float4(a0x + a1x, a0y + a1y, a0z + a1z, a0w + a1w);

    WAIT_ASYNC();      // async LDS writes landed
    __syncthreads();   // tiles ready (DScnt covers the xs store)

    // --- WMMA: D(16x16) += A(16x4, fp32) x B(4x16, fp32), 16 steps ---
    const float* xrow = sm.xs[fm];
    const float* wrow = sm.w1s[wv * 16 + fm];
    #pragma unroll
    for (int kk = 0; kk < CK; kk += 4) {
      const v2f a = *(const v2f*)(xrow + kk + hi2);
      const v2f b = *(const v2f*)(wrow + kk + hi2);
      cf = __builtin_amdgcn_wmma_f32_16x16x4_f32(
          /*neg_a=*/false, a, /*neg_b=*/false, b,
          /*c_mod=*/(short)0, cf, /*reuse_a=*/false, /*reuse_b=*/false);
    }
  }

  // --- epilogue: bias + ReLU + W2 scale, write partials for cross-wave reduce ---
  {
    const int   j   = wv * 16 + fm;
    const float b1j = b1[j];
    const float w2j = W2[j];
    #pragma unroll
    for (int v = 0; v < 8; ++v) {
      float h = cf[v] + b1j;
      h = h > 0.f ? h : 0.f;
      sm.part[wv][lane][v] = h * w2j;
    }
  }
  __syncthreads();

  // Fragment layout: VGPR v holds row M=v (lanes 0-15) / M=v+8 (lanes 16-31), col N=lane%16
  if (t < TM) {
    const int vr = t & 7;
    const int lb = (t >> 3) * 16;
    float s = b2[0];
    for (int w = 0; w < 8; ++w)
      #pragma unroll
      for (int n = 0; n < 16; ++n)
        s += sm.part[w][lb + n][vr];
    out[gr0 + t] = s;
  }
}

extern "C" void kernel_launch(void* const* d_in, const int* in_sizes, int n_in,
                              void* d_out, int out_size, void* d_ws, size_t ws_size,
                              hipStream_t stream) {
  const int*   white = (const int*)  d_in[0];
  const int*   black = (const int*)  d_in[1];
  const int*   stm   = (const int*)  d_in[2];
  const float* emb   = (const float*)d_in[3];
  const float* W1    = (const float*)d_in[4];
  const float* b1    = (const float*)d_in[5];
  const float* W2    = (const float*)d_in[6];
  const float* b2    = (const float*)d_in[7];
  float*       out   = (float*)d_out;

  const int B = in_sizes[2];          // 4096
  nnue_fused_kernel<<<B / TM, 256, 0, stream>>>(white, black, stm, emb,
                                                W1, b1, W2, b2, out);
}